// Model_n_25443386261716
// MI455X (gfx1250) — compile-verified
//
#include <hip/hip_runtime.h>

typedef __attribute__((ext_vector_type(2))) float v2f;
typedef __attribute__((ext_vector_type(8))) float v8f;
typedef __attribute__((ext_vector_type(4))) unsigned int u32x4;
typedef __attribute__((ext_vector_type(4))) int          i32x4;
typedef __attribute__((ext_vector_type(8))) int          i32x8;

static constexpr int BATCH   = 128;
static constexpr int TSTEPS  = 60;
static constexpr int IN_CH   = 2;
static constexpr int HW      = 32;
static constexpr int KSZ     = 6;
static constexpr int OUT_HW  = 27;            // 32 - 6 + 1
static constexpr int NK      = 16;
static constexpr int POS     = OUT_HW*OUT_HW; // 729
static constexpr int FLAT    = POS*NK;        // 11664
static constexpr int KTOT    = IN_CH*KSZ*KSZ; // 72
static constexpr int RNN     = 16;
static constexpr int IMGS    = BATCH*TSTEPS;  // 7680
static constexpr int XSZ     = IN_CH*HW*HW;   // 2048 floats / frame

// flat index k = ic*36 + kh*6 + kw  ->  LDS offset into [2][32][32] frame
__host__ __device__ constexpr int off_k(int k) {
    return (k/36)*(HW*HW) + ((k%36)/6)*HW + ((k%36)%6);
}

// ---------------------------------------------------------------------------
// TDM: DMA one fp32 frame (2048 elems) from global to LDS offset 0.
// D# per CDNA5 ISA (1-row 2D tile, 4B elems).
// ---------------------------------------------------------------------------
__device__ __forceinline__ void tdm_load_frame_to_lds0(const float* gsrc)
{
    const unsigned long long ga = (unsigned long long)(const void*)gsrc;
    u32x4 g0;
    g0[0] = 1u;                                   // count=1, is_restore=0
    g0[1] = 0u;                                   // lds_addr = 0 (xs base)
    g0[2] = (unsigned int)ga;                     // global_addr[31:0]
    g0[3] = ((unsigned int)(ga >> 32) & 0x01FFFFFFu) | (2u << 30); // addr hi | type=2

    i32x8 g1 = {};
    g1[0] = (2 << 16);                            // data_size = 4 bytes
    g1[1] = (XSZ & 0xFFFF) << 16;                 // tensor_dim0[15:0] = 2048
    g1[2] = (XSZ >> 16) | (1 << 16);              // tensor_dim0 hi | tensor_dim1 = 1
    g1[3] = (XSZ & 0xFFFF) << 16;                 // tile_dim0 = 2048
    g1[4] = 1;                                    // tile_dim1 = 1
    g1[5] = XSZ;                                  // tensor_dim0_stride = 2048
    g1[6] = 0;
    g1[7] = 0;

#if defined(__clang_major__) && (__clang_major__ >= 23)
    i32x4 gz4 = {};
    i32x8 gz8 = {};
    __builtin_amdgcn_tensor_load_to_lds(g0, g1, gz4, gz4, gz8, 0);
#else
    i32x4 gz4 = {};
    __builtin_amdgcn_tensor_load_to_lds(g0, g1, gz4, gz4, 0);
#endif
    __builtin_amdgcn_s_wait_tensorcnt(0);
}

// ---------------------------------------------------------------------------
// Kernel 1: conv(6x6,VALID) + bias, writes out_conv and out_flat (=relu).
// One block per frame. D[16ch x 16pos] = W[16x72] x P[72x16] via 18 chained
// v_wmma_f32_16x16x4_f32. Frame staged in LDS (8KB) by the Tensor Data Mover.
// ---------------------------------------------------------------------------
__global__ __launch_bounds__(256)
void conv_wmma_kernel(const float* __restrict__ x,
                      const float* __restrict__ cw,
                      const float* __restrict__ cb,
                      float* __restrict__ out_conv,
                      float* __restrict__ out_flat)
{
    __shared__ float xs[XSZ];                     // at LDS offset 0
    const int img  = blockIdx.x;
    const int tid  = threadIdx.x;
    const int wv   = tid >> 5;
    const int lane = tid & 31;
    const int m    = lane & 15;   // A: channel row / B,D: column
    const int hi   = lane >> 4;   // upper lane half -> K+2 / M+8

    if (wv == 0) {                                // wave-uniform branch
        tdm_load_frame_to_lds0(x + (size_t)img * XSZ);
    }

    // A = conv weights, 16x72, 18 chunks of K=4 (v2f per lane) — overlaps DMA
    v2f a[18];
#pragma unroll
    for (int c = 0; c < 18; ++c) {
        const float* p = cw + m*KTOT + 4*c + 2*hi;  // 8B aligned
        a[c] = *(const v2f*)p;
    }
    float bias[8];
#pragma unroll
    for (int v = 0; v < 8; ++v) bias[v] = cb[v + 8*hi];

    __syncthreads();                              // publish LDS frame

    // 46 tiles of 16 positions (729 padded to 736), waves stride over tiles
    for (int tile = wv; tile < 46; tile += 8) {
        const int pos  = tile*16 + m;
        const int cpos = pos < POS ? pos : POS-1;     // clamp for LDS reads
        const int py   = cpos / OUT_HW;
        const int px   = cpos - py*OUT_HW;
        const int pb   = py*HW + px;

        v8f acc = {};
#pragma unroll
        for (int c = 0; c < 18; ++c) {
            // lane supplies B[k0+2*hi + {0,1}][pos]; offsets are immediates
            const int o0 = hi ? off_k(4*c + 2) : off_k(4*c + 0);
            const int o1 = hi ? off_k(4*c + 3) : off_k(4*c + 1);
            v2f bb = { xs[o0 + pb], xs[o1 + pb] };
            acc = __builtin_amdgcn_wmma_f32_16x16x4_f32(
                    false, a[c], false, bb, (short)0, acc, false, false);
        }

        if (pos < POS) {
            const size_t base = (size_t)img * FLAT + pos;
#pragma unroll
            for (int v = 0; v < 8; ++v) {
                const int ch = v + 8*hi;              // D: VGPR v -> row v(+8)
                const float val = acc[v] + bias[v];
                out_conv[base + (size_t)ch*POS] = val;
                out_flat[base + (size_t)ch*POS] = fmaxf(val, 0.0f);
            }
        }
    }
}

// ---------------------------------------------------------------------------
// Kernel 2: z[img][u] = out_flat[img][:] @ w_ih[u][:] + b_ih[u]
// Block = 16 frames; 8 waves split K (2916 chunks of 4), LDS reduction.
// ---------------------------------------------------------------------------
__global__ __launch_bounds__(256)
void proj_wmma_kernel(const float* __restrict__ flat,
                      const float* __restrict__ w_ih,
                      const float* __restrict__ b_ih,
                      float* __restrict__ z)
{
    __shared__ float red[8][32][8];
    const int tid  = threadIdx.x;
    const int wv   = tid >> 5;
    const int lane = tid & 31;
    const int m    = lane & 15;
    const int hi   = lane >> 4;
    const int img0 = blockIdx.x * 16;

    const float* arow = flat + (size_t)(img0 + m) * FLAT + 2*hi; // A: M=frame
    const float* brow = w_ih + (size_t)m * FLAT + 2*hi;          // B: N=unit

    v8f acc = {};
    for (int c = wv; c < FLAT/4; c += 8) {
        __builtin_prefetch(arow + 4*(c + 8), 0, 0);  // global_prefetch_b8
        v2f av = *(const v2f*)(arow + 4*c);
        v2f bv = *(const v2f*)(brow + 4*c);
        acc = __builtin_amdgcn_wmma_f32_16x16x4_f32(
                false, av, false, bv, (short)0, acc, false, false);
    }
#pragma unroll
    for (int v = 0; v < 8; ++v) red[wv][lane][v] = acc[v];
    __syncthreads();

    // 256 threads: one (lane,vgpr) cell each, sum across 8 waves
    const int l = tid >> 3;
    const int v = tid & 7;
    float s = 0.0f;
#pragma unroll
    for (int w = 0; w < 8; ++w) s += red[w][l][v];
    const int mm  = l & 15;            // unit
    const int hh  = l >> 4;
    const int img = img0 + v + 8*hh;   // D row = frame
    z[(size_t)img*RNN + mm] = s + b_ih[mm];
}

// ---------------------------------------------------------------------------
// Kernel 3: h_t = relu(z_t + h_{t-1} @ w_hh^T + b_hh); fc fused.
// One wave per batch element (128 waves), h shared via LDS per wave.
// ---------------------------------------------------------------------------
__global__ __launch_bounds__(512)
void rnn_kernel(const float* __restrict__ z,
                const float* __restrict__ w_hh,
                const float* __restrict__ b_hh,
                const float* __restrict__ fc_w,
                const float* __restrict__ fc_b,
                float* __restrict__ out_fc,
                float* __restrict__ out_rnn)
{
    __shared__ float hs[16][RNN];
    const int tid  = threadIdx.x;
    const int wv   = tid >> 5;                  // wave in block: 0..15
    const int lane = tid & 31;
    const int b    = blockIdx.x * 16 + wv;      // batch element
    const int u    = lane & 15;                 // unit (lanes 16..31 shadow)

    float wrow[RNN];
#pragma unroll
    for (int v = 0; v < RNN; ++v) wrow[v] = w_hh[u*RNN + v];
    const float bh = b_hh[u];

    if (lane < RNN) hs[wv][u] = 0.0f;

    for (int t = 0; t < TSTEPS; ++t) {
        const size_t idx = (size_t)b * TSTEPS + t;
        float hw = 0.0f;
#pragma unroll
        for (int v = 0; v < RNN; ++v) hw += hs[wv][v] * wrow[v];
        const float zt = z[idx*RNN + u];
        const float h  = fmaxf(zt + hw + bh, 0.0f);
        if (lane < RNN) { hs[wv][u] = h; out_rnn[idx*RNN + u] = h; }
        if (lane < 2) {
            float s = fc_b[lane];
#pragma unroll
            for (int v = 0; v < RNN; ++v) s += hs[wv][v] * fc_w[lane*RNN + v];
            out_fc[idx*2 + lane] = s;
        }
    }
}

// ---------------------------------------------------------------------------
extern "C" void kernel_launch(void* const* d_in, const int* in_sizes, int n_in,
                              void* d_out, int out_size, void* d_ws, size_t ws_size,
                              hipStream_t stream)
{
    const float* x      = (const float*)d_in[0];
    const float* conv_w = (const float*)d_in[1];
    const float* conv_b = (const float*)d_in[2];
    const float* w_ih   = (const float*)d_in[3];
    const float* w_hh   = (const float*)d_in[4];
    const float* b_ih   = (const float*)d_in[5];
    const float* b_hh   = (const float*)d_in[6];
    const float* fc_w   = (const float*)d_in[7];
    const float* fc_b   = (const float*)d_in[8];

    float* out      = (float*)d_out;
    float* out_fc   = out;                                   // [B,T,2]
    float* out_conv = out_fc   + (size_t)IMGS * 2;           // [B,T,16,27,27]
    float* out_flat = out_conv + (size_t)IMGS * FLAT;        // [B,T,11664]
    float* out_rnn  = out_flat + (size_t)IMGS * FLAT;        // [B,T,16]

    float* z = (float*)d_ws;                                 // [IMGS,16]

    conv_wmma_kernel<<<IMGS,    256, 0, stream>>>(x, conv_w, conv_b, out_conv, out_flat);
    proj_wmma_kernel<<<IMGS/16, 256, 0, stream>>>(out_flat, w_ih, b_ih, z);
    rnn_kernel<<<BATCH/16,      512, 0, stream>>>(z, w_hh, b_hh, fc_w, fc_b, out_fc, out_rnn);
}